// GCN_35107062677929
// MI455X (gfx1250) — compile-verified
//
#include <hip/hip_runtime.h>
#include <hip/hip_bf16.h>

// CDNA5 / gfx1250 GCN forward pass.
// Memory-bound on edge gather/scatter (~3 GB @ 23.3 TB/s); GEMMs (~9 GFLOP)
// are done with V_WMMA_F32_16X16X4_F32 to keep fp32 numerics while using the
// matrix pipe. Wave32 everywhere.

typedef __attribute__((ext_vector_type(2))) float v2f;
typedef __attribute__((ext_vector_type(8))) float v8f;

#define LRELU(v) ((v) > 0.0f ? (v) : 0.01f * (v))

// ---------------------------------------------------------------------------
// Degree / normalization helpers
// ---------------------------------------------------------------------------
__global__ void __launch_bounds__(256) fill_ones_k(float* p, int n) {
    int i = blockIdx.x * 256 + threadIdx.x;
    if (i < n) p[i] = 1.0f;  // self-loop contributes 1 to every node's degree
}

__global__ void __launch_bounds__(256) count_deg_k(const int* __restrict__ dst,
                                                   float* __restrict__ deg, int E) {
    int i = blockIdx.x * 256 + threadIdx.x;
    if (i < E) atomicAdd(&deg[dst[i]], 1.0f);
}

__global__ void __launch_bounds__(256) rsqrt_k(float* p, int n) {
    int i = blockIdx.x * 256 + threadIdx.x;
    if (i < n) p[i] = rsqrtf(p[i]);  // deg >= 1 always (self-loops)
}

// ---------------------------------------------------------------------------
// GEMM: C[M,Nout] = act(A[M,K] @ W[K,Nout] + bias)
// One wave per 16x16 C tile, 8 waves / block share a K x 16 panel of W in LDS.
// A fragment (16x4 f32): lane l (l<16) holds row M=l, K={k0,k0+1};
//                        lane l (l>=16) holds row M=l-16, K={k0+2,k0+3}.
// B fragment (4x16 f32): low lanes rows K={k0,k0+1}, high lanes K={k0+2,k0+3}.
// C/D (16x16 f32): VGPR r -> M=r (lanes 0-15), M=r+8 (lanes 16-31), N=lane&15.
// ---------------------------------------------------------------------------
__global__ void __launch_bounds__(256)
gemm_wmma_f32(const float* __restrict__ A, const float* __restrict__ W,
              const float* __restrict__ bias, float* __restrict__ C,
              int M, int K, int Nout, int act) {
    __shared__ float Ws[256 * 16];  // K <= 256
    const int tid  = threadIdx.x;
    const int lane = tid & 31;
    const int wave = tid >> 5;
    const int n0   = blockIdx.y * 16;

    // Cooperative preload of W[:, n0:n0+16] into LDS.
    for (int idx = tid; idx < K * 16; idx += 256) {
        int k = idx >> 4, n = idx & 15;
        Ws[idx] = W[(size_t)k * Nout + n0 + n];
    }
    __syncthreads();

    const int mTile = blockIdx.x * 8 + wave;
    if (mTile * 16 >= M) return;  // wave-uniform: EXEC stays all-ones for WMMA

    const int hi  = lane >> 4;        // 0: K pair {0,1}; 1: K pair {2,3}
    const int l15 = lane & 15;
    const float* __restrict__ Arow = A + (size_t)(mTile * 16 + l15) * K;

    v8f acc = {};
    for (int k0 = 0; k0 < K; k0 += 4) {
        v2f a, b;
        a.x = Arow[k0 + 2 * hi];
        a.y = Arow[k0 + 2 * hi + 1];
        b.x = Ws[(k0 + 2 * hi) * 16 + l15];
        b.y = Ws[(k0 + 2 * hi + 1) * 16 + l15];
        acc = __builtin_amdgcn_wmma_f32_16x16x4_f32(
            /*neg_a=*/false, a, /*neg_b=*/false, b,
            /*c_mod=*/(short)0, acc, /*reuse_a=*/false, /*reuse_b=*/false);
    }

    const int n = n0 + l15;
    const float bv = bias ? bias[n] : 0.0f;
#pragma unroll
    for (int r = 0; r < 8; ++r) {
        int row = mTile * 16 + r + 8 * hi;
        float v = acc[r] + bv;
        if (act) v = LRELU(v);
        C[(size_t)row * Nout + n] = v;
    }
}

// ---------------------------------------------------------------------------
// out[i,c] = t[i,c] * dis[i]^2 + bias[c]   (self-loop message + bias init)
// ---------------------------------------------------------------------------
__global__ void __launch_bounds__(256)
agg_init_k(const float* __restrict__ t, const float* __restrict__ dis,
           const float* __restrict__ bias, float* __restrict__ out,
           int N, int C) {
    int i = blockIdx.x * 256 + threadIdx.x;
    if (i >= N * C) return;
    int node = i / C, c = i - node * C;
    float w = dis[node];
    out[i] = t[i] * (w * w) + bias[c];
}

// ---------------------------------------------------------------------------
// Edge scatter: out[dst] += t[src] * dis[src]*dis[dst]
// One wave per edge; float4 gathers, global_atomic_add_f32 scatters.
// ---------------------------------------------------------------------------
__global__ void __launch_bounds__(256)
scatter_k(const float* __restrict__ t, const float* __restrict__ dis,
          const int* __restrict__ src, const int* __restrict__ dst,
          float* __restrict__ out, int E, int C) {
    const int lane = threadIdx.x & 31;
    const int e    = blockIdx.x * 8 + (threadIdx.x >> 5);
    if (e >= E) return;
    const int s = src[e], d = dst[e];
    const float norm = dis[s] * dis[d];
    const float4* __restrict__ ts = (const float4*)(t + (size_t)s * C);
    float* __restrict__ od = out + (size_t)d * C;
    for (int c4 = lane; c4 < (C >> 2); c4 += 32) {
        float4 v = ts[c4];
        int c = c4 << 2;
        atomicAdd(&od[c + 0], v.x * norm);
        atomicAdd(&od[c + 1], v.y * norm);
        atomicAdd(&od[c + 2], v.z * norm);
        atomicAdd(&od[c + 3], v.w * norm);
    }
}

__global__ void __launch_bounds__(256) lrelu_k(float* p, int n) {
    int i = blockIdx.x * 256 + threadIdx.x;
    if (i < n) { float v = p[i]; p[i] = LRELU(v); }
}

// ---------------------------------------------------------------------------
// out[n] = h[n,:64] . w + b   — one wave per node, shfl_xor wave32 reduction
// ---------------------------------------------------------------------------
__global__ void __launch_bounds__(256)
rowdot64_k(const float* __restrict__ h, const float* __restrict__ w,
           const float* __restrict__ b, float* __restrict__ out, int N) {
    const int lane = threadIdx.x & 31;
    const int node = blockIdx.x * 8 + (threadIdx.x >> 5);
    if (node >= N) return;
    const float2 hv = ((const float2*)(h + (size_t)node * 64))[lane];
    const float2 wv = ((const float2*)w)[lane];
    float p = hv.x * wv.x + hv.y * wv.y;
#pragma unroll
    for (int off = 16; off > 0; off >>= 1) p += __shfl_xor(p, off, 32);
    if (lane == 0) out[node] = p + b[0];
}

// ---------------------------------------------------------------------------
// Host-side launch
// ---------------------------------------------------------------------------
static inline void launch_gemm(const float* A, const float* W, const float* bias,
                               float* C, int M, int K, int Nout, int act,
                               hipStream_t s) {
    int mTiles = (M + 15) / 16;
    dim3 grid((mTiles + 7) / 8, Nout / 16);
    gemm_wmma_f32<<<grid, 256, 0, s>>>(A, W, bias, C, M, K, Nout, act);
}

extern "C" void kernel_launch(void* const* d_in, const int* in_sizes, int n_in,
                              void* d_out, int out_size, void* d_ws, size_t ws_size,
                              hipStream_t stream) {
    const float* x     = (const float*)d_in[0];
    const int*   ei    = (const int*)d_in[1];
    const float* W_in  = (const float*)d_in[2];
    const float* b_in  = (const float*)d_in[3];
    const float* W1    = (const float*)d_in[4];
    const float* b1    = (const float*)d_in[5];
    const float* W2    = (const float*)d_in[6];
    const float* b2    = (const float*)d_in[7];
    const float* W3    = (const float*)d_in[8];
    const float* b3    = (const float*)d_in[9];
    const float* W_out = (const float*)d_in[10];
    const float* b_out = (const float*)d_in[11];

    const int D = 128;
    const int N = in_sizes[0] / D;   // 50000 (divisible by 16)
    const int E = in_sizes[1] / 2;   // 800000
    const int* src = ei;
    const int* dst = ei + E;

    // Workspace carve-up (fp32): dis[N] | bufA[N*128] | bufB[N*256] | bufC[N*256]
    float* dis  = (float*)d_ws;
    float* bufA = dis + (((size_t)N + 255) & ~(size_t)255);
    float* bufB = bufA + (size_t)N * 128;
    float* bufC = bufB + (size_t)N * 256;

    const int BN  = (N + 255) / 256;
    const int BE  = (E + 255) / 256;
    const int BEw = (E + 7) / 8;

    // Normalization: dis = rsqrt(in_degree + 1)
    fill_ones_k<<<BN, 256, 0, stream>>>(dis, N);
    count_deg_k<<<BE, 256, 0, stream>>>(dst, dis, E);
    rsqrt_k<<<BN, 256, 0, stream>>>(dis, N);

    // h0 = lrelu(x @ W_in + b_in)           -> bufA [N,128]
    launch_gemm(x, W_in, b_in, bufA, N, 128, 128, 1, stream);

    // conv1: 128 -> 256                      h1 -> bufC
    launch_gemm(bufA, W1, nullptr, bufB, N, 128, 256, 0, stream);
    agg_init_k<<<(N * 256 + 255) / 256, 256, 0, stream>>>(bufB, dis, b1, bufC, N, 256);
    scatter_k<<<BEw, 256, 0, stream>>>(bufB, dis, src, dst, bufC, E, 256);
    lrelu_k<<<(N * 256 + 255) / 256, 256, 0, stream>>>(bufC, N * 256);

    // conv2: 256 -> 128                      h2 -> bufB
    launch_gemm(bufC, W2, nullptr, bufA, N, 256, 128, 0, stream);
    agg_init_k<<<(N * 128 + 255) / 256, 256, 0, stream>>>(bufA, dis, b2, bufB, N, 128);
    scatter_k<<<BEw, 256, 0, stream>>>(bufA, dis, src, dst, bufB, E, 128);
    lrelu_k<<<(N * 128 + 255) / 256, 256, 0, stream>>>(bufB, N * 128);

    // conv3: 128 -> 64                       h3 -> bufA
    launch_gemm(bufB, W3, nullptr, bufC, N, 128, 64, 0, stream);
    agg_init_k<<<(N * 64 + 255) / 256, 256, 0, stream>>>(bufC, dis, b3, bufA, N, 64);
    scatter_k<<<BEw, 256, 0, stream>>>(bufC, dis, src, dst, bufA, E, 64);
    lrelu_k<<<(N * 64 + 255) / 256, 256, 0, stream>>>(bufA, N * 64);

    // out = h3 @ W_out + b_out               -> d_out [N,1]
    rowdot64_k<<<(N + 7) / 8, 256, 0, stream>>>(bufA, W_out, b_out, (float*)d_out, N);
}